// MoLoRa_3109556322595
// MI455X (gfx1250) — compile-verified
//
#include <hip/hip_runtime.h>

// ---- problem constants (match reference) ----
static constexpr int   Dd      = 2048;   // hidden dim
static constexpr int   Ee      = 8;      // experts
static constexpr int   Rr      = 4;      // LoRA rank
static constexpr int   ER      = Ee * Rr;      // 32 combined (e,r)
static constexpr int   TILE_M  = 16;     // tokens per wave (WMMA M)
static constexpr int   WAVES   = 4;      // waves per block
static constexpr int   DCHUNK  = 128;    // x staging chunk (floats)
static constexpr int   XST     = DCHUNK + 4; // LDS x-tile row stride (even, 16B rows, conflict-free)
static constexpr int   AXST    = 49;     // LDS phase-1 spill row stride
static constexpr int   WST     = 34;     // LDS w-tile row stride (even -> 8B-aligned float2 reads)
static constexpr float SCALE   = 4.0f;   // ALPHA / RANK

typedef __attribute__((ext_vector_type(2))) float v2f;
typedef __attribute__((ext_vector_type(4))) float v4f;
typedef __attribute__((ext_vector_type(8))) float v8f;

// D = A(16x4 f32) * B(4x16 f32) + C(16x16 f32)
#define WMMA_F32(a, b, c) \
  __builtin_amdgcn_wmma_f32_16x16x4_f32(false, (a), false, (b), (short)0, (c), false, false)

__global__ __launch_bounds__(WAVES * 32)
void molora_fused(const float* __restrict__ x,    // [T, D]
                  const float* __restrict__ lA,   // [E, D, R]
                  const float* __restrict__ lB,   // [E, R, D] == [ER, D]
                  const float* __restrict__ rw,   // [D, E]
                  float* __restrict__ out)        // [T, D]
{
  __shared__ float xs [WAVES][TILE_M * XST];   // x tile stage
  __shared__ float axl[WAVES][TILE_M * AXST];  // phase-1 results: 32 ax cols + 8 logits + pad
  __shared__ float wl [WAVES][TILE_M * WST];   // router-weighted w tile (phase-2 A operand)

  const int lane = threadIdx.x & 31;
  const int wave = threadIdx.x >> 5;
  const int n16  = lane & 15;
  const int hi   = lane >> 4;                  // 0: K rows {0,1}, 1: K rows {2,3}
  const int tokBase = (blockIdx.x * WAVES + wave) * TILE_M;

  float* xw  = xs [wave];
  float* axw = axl[wave];
  float* ww  = wl [wave];

  // per-lane column constants for the lora_A B-operand:
  // column er = e*4 + r ; lA layout [E][D][R] -> idx = e*D*R + d*R + r
  const int e0  = n16 >> 2;
  const int r0  = n16 & 3;
  const int cb0 = (e0      * Dd) * Rr + r0;    // experts 0..3  (cols  0..15)
  const int cb1 = ((e0 + 4) * Dd) * Rr + r0;   // experts 4..7  (cols 16..31)
  // router B-operand: lanes 8..15 duplicate columns 0..7 unconditionally.
  // The duplicate logits land in phase-1 output columns 40..47, which are
  // never read by the softmax stage -- no masking, no EXEC divergence.
  const int rcol = n16 & (Ee - 1);

  v8f acc0 = {}, acc1 = {}, acc2 = {};         // [16 x 16] f32 each

  // ================= phase 1: ax = x @ A  and  logits = x @ Wr =================
  for (int d0 = 0; d0 < Dd; d0 += DCHUNK) {
    // stage x[16 x DCHUNK] into LDS, fully coalesced (one float4 per lane per row)
    #pragma unroll 4
    for (int m = 0; m < TILE_M; ++m) {
      v4f v = *(const v4f*)(x + (tokBase + m) * Dd + d0 + lane * 4);
      *(v4f*)(xw + m * XST + lane * 4) = v;
    }
    if (d0 + DCHUNK < Dd)
      __builtin_prefetch(x + tokBase * Dd + d0 + DCHUNK, 0, 3); // near-scope global_prefetch
    __asm__ volatile("s_wait_dscnt 0" ::: "memory");

    #pragma unroll 4
    for (int kk = 0; kk < DCHUNK; kk += 4) {
      const int dk = d0 + kk + 2 * hi;
      // shared A operand (x tile), ISA 16x4 f32 layout: lane m holds K = {2hi, 2hi+1}
      v2f a = *(const v2f*)(xw + n16 * XST + kk + 2 * hi);

      v2f b0; b0.x = lA[cb0 + dk * Rr];        b0.y = lA[cb0 + dk * Rr + Rr];
      acc0 = WMMA_F32(a, b0, acc0);

      v2f b1; b1.x = lA[cb1 + dk * Rr];        b1.y = lA[cb1 + dk * Rr + Rr];
      acc1 = WMMA_F32(a, b1, acc1);

      v2f b2;                                   // router logits (cols 8..15 = ignored dups)
      b2.x = rw[dk       * Ee + rcol];
      b2.y = rw[(dk + 1) * Ee + rcol];
      acc2 = WMMA_F32(a, b2, acc2);
    }
  }

  // spill phase-1 accumulators (documented C layout: VGPR i -> row i + 8*hi, lane -> col)
  #pragma unroll
  for (int i = 0; i < 8; ++i) {
    const int row = (i + 8 * hi) * AXST;
    axw[row +        n16] = acc0[i];
    axw[row + 16 +   n16] = acc1[i];
    axw[row + 32 +   n16] = acc2[i];
  }
  __asm__ volatile("s_wait_dscnt 0" ::: "memory");

  // ================= softmax + router-weighted scaling (one token per lane) =================
  if (lane < TILE_M) {
    const float* row = axw + lane * AXST;
    float mx = row[32];
    #pragma unroll
    for (int e = 1; e < Ee; ++e) mx = fmaxf(mx, row[32 + e]);
    float p[Ee], s = 0.0f;
    #pragma unroll
    for (int e = 0; e < Ee; ++e) { p[e] = __expf(row[32 + e] - mx); s += p[e]; }
    const float inv = SCALE / s;              // fold ALPHA/RANK and 1/sum into w
    #pragma unroll
    for (int er = 0; er < ER; ++er)
      ww[lane * WST + er] = row[er] * p[er >> 2] * inv;
  }
  __asm__ volatile("s_wait_dscnt 0" ::: "memory");

  // hoist the whole w[16 x 32] tile into registers as 8 WMMA A-operands
  v2f aop[8];
  #pragma unroll
  for (int kk = 0; kk < 8; ++kk)
    aop[kk] = *(const v2f*)(ww + n16 * WST + 4 * kk + 2 * hi);

  // ================= phase 2: out[16 x D] = w[16 x 32] @ lora_B[32 x D] =================
  for (int g = 0; g < Dd; g += 16) {
    v8f acc = {};
    #pragma unroll
    for (int kk = 0; kk < 8; ++kk) {
      const int er = 4 * kk + 2 * hi;          // B operand 4x16 layout: rows {er, er+1}
      v2f b;
      b.x = lB[er       * Dd + g + n16];
      b.y = lB[(er + 1) * Dd + g + n16];
      acc = WMMA_F32(aop[kk], b, acc);
    }
    #pragma unroll
    for (int i = 0; i < 8; ++i)
      out[(tokBase + i + 8 * hi) * Dd + g + n16] = acc[i];
  }
}

extern "C" void kernel_launch(void* const* d_in, const int* in_sizes, int n_in,
                              void* d_out, int out_size, void* d_ws, size_t ws_size,
                              hipStream_t stream) {
  const float* x  = (const float*)d_in[0];   // [B,S,D] fp32
  const float* lA = (const float*)d_in[1];   // [E,D,R] fp32
  const float* lB = (const float*)d_in[2];   // [E,R,D] fp32
  const float* rw = (const float*)d_in[3];   // [D,E]   fp32
  float* out = (float*)d_out;

  const int T    = in_sizes[0] / Dd;               // 8192 tokens
  const int grid = T / (TILE_M * WAVES);           // 128 blocks
  molora_fused<<<grid, WAVES * 32, 0, stream>>>(x, lA, lB, rw, out);
}